// FCOSPostProcessor_30528627540293
// MI455X (gfx1250) — compile-verified
//
#include <hip/hip_runtime.h>
#include <math.h>

// FCOS post-processing for MI455X (gfx1250, wave32).
// B=16 images, N=50000 locations, C=80 classes, K=1000 pre-NMS, M=100 outputs.

#define BB    16
#define NN    50000
#define CC    80
#define KK    1000
#define KPADD 1024
#define MM    100
#define CAPC  4096
#define NBINS 4096
#define LOGIT_T (-2.9444389791664403f)   // log(0.05/0.95): sigmoid(x)>0.05
#define NMS_T 0.6f
#define MIN_SZ 0.01f
#define SCORE_T 0.05f

typedef __attribute__((ext_vector_type(2))) float v2f;
typedef __attribute__((ext_vector_type(8))) float v8f;

// fast sigmoid: v_exp_f32 + v_rcp_f32 (no IEEE divide sequence)
__device__ __forceinline__ float sigm(float x) {
  return __builtin_amdgcn_rcpf(1.f + __expf(-x));
}

// ---------------- workspace zeroing (atomics accumulate, must reset per call) ---
__global__ void init_ws(int* hist, int* cnt, unsigned* supp) {
  int tid = blockIdx.x * blockDim.x + threadIdx.x;
  int stride = gridDim.x * blockDim.x;
  for (int t = tid; t < BB * NBINS; t += stride) hist[t] = 0;
  for (int t = tid; t < BB; t += stride) cnt[t] = 0;
  for (int t = tid; t < BB * KPADD * (KPADD / 32); t += stride) supp[t] = 0u;
}

// ---------------- precompute sigmoid(centerness) once (was recomputed 80x) -----
__global__ void cness_kernel(const float* __restrict__ cness,
                             float* __restrict__ cnsig) {
  int tid = blockIdx.x * blockDim.x + threadIdx.x;
  int stride = gridDim.x * blockDim.x;
  for (int t = tid; t < BB * NN; t += stride) cnsig[t] = sigm(cness[t]);
}

// ---------------- pass 1: per-image score histogram (bandwidth bound) ----------
// float4 loads: C=80 divisible by 4, so a 4-aligned group shares one location n.
__global__ void hist_kernel(const float* __restrict__ cls,
                            const float* __restrict__ cnsig,
                            int* __restrict__ hist) {
  int b = blockIdx.y;
  const float4* clsb = (const float4*)(cls + (size_t)b * NN * CC);
  const float* cnb = cnsig + (size_t)b * NN;
  __shared__ int lh[NBINS];
  for (int t = threadIdx.x; t < NBINS; t += blockDim.x) lh[t] = 0;
  __syncthreads();
  const int total4 = NN * CC / 4;
  const int stride = gridDim.x * blockDim.x;
  for (int e = blockIdx.x * blockDim.x + threadIdx.x; e < total4; e += stride) {
    __builtin_prefetch(clsb + e + stride, 0, 0);   // global_prefetch_b8
    float4 x = clsb[e];
    float cn = cnb[e / (CC / 4)];
    float xv[4] = {x.x, x.y, x.z, x.w};
#pragma unroll
    for (int q = 0; q < 4; ++q) {
      if (xv[q] > LOGIT_T) {
        float s = sigm(xv[q]) * cn;
        int bin = (int)(s * (float)NBINS);
        bin = bin < 0 ? 0 : (bin > NBINS - 1 ? NBINS - 1 : bin);
        atomicAdd(&lh[bin], 1);
      }
    }
  }
  __syncthreads();
  for (int t = threadIdx.x; t < NBINS; t += blockDim.x) {
    int v = lh[t];
    if (v) atomicAdd(&hist[b * NBINS + t], v);
  }
}

// ---------------- per-image top-K threshold bin --------------------------------
__global__ void thresh_kernel(const int* __restrict__ hist, int* __restrict__ Tbin) {
  int b = blockIdx.x;
  if (threadIdx.x == 0) {
    int cum = 0, T = 0;
    for (int bin = NBINS - 1; bin >= 0; --bin) {
      cum += hist[b * NBINS + bin];
      if (cum >= KK) { T = bin; break; }
    }
    Tbin[b] = T;
  }
}

// ---------------- pass 2: collect candidates >= threshold ----------------------
__global__ void collect_kernel(const float* __restrict__ cls,
                               const float* __restrict__ cnsig,
                               const int* __restrict__ Tbin,
                               int* __restrict__ cnt,
                               float* __restrict__ cscore,
                               int* __restrict__ cidx) {
  int b = blockIdx.y;
  const float4* clsb = (const float4*)(cls + (size_t)b * NN * CC);
  const float* cnb = cnsig + (size_t)b * NN;
  float tlow = (float)Tbin[b] / (float)NBINS;
  const int total4 = NN * CC / 4;
  const int stride = gridDim.x * blockDim.x;
  for (int e = blockIdx.x * blockDim.x + threadIdx.x; e < total4; e += stride) {
    __builtin_prefetch(clsb + e + stride, 0, 0);
    float4 x = clsb[e];
    float cn = cnb[e / (CC / 4)];
    float xv[4] = {x.x, x.y, x.z, x.w};
#pragma unroll
    for (int q = 0; q < 4; ++q) {
      if (xv[q] > LOGIT_T) {
        float s = sigm(xv[q]) * cn;
        if (s >= tlow) {
          int pos = atomicAdd(&cnt[b], 1);
          if (pos < CAPC) { cscore[b * CAPC + pos] = s; cidx[b * CAPC + pos] = e * 4 + q; }
        }
      }
    }
  }
}

// ---------------- per-image: bitonic sort (LDS) + box decode -------------------
__global__ __launch_bounds__(1024) void sort_decode_kernel(
    const int* __restrict__ cnt, const float* __restrict__ cscore,
    const int* __restrict__ cidx, const float* __restrict__ loc,
    const float* __restrict__ reg, const int* __restrict__ img,
    float* __restrict__ boxes, float* __restrict__ ss, int* __restrict__ sv,
    int* __restrict__ lab, float* __restrict__ areas) {
  __shared__ float ks[CAPC];
  __shared__ int   kid[CAPC];
  int b = blockIdx.x;
  int tid = threadIdx.x;
  int c = cnt[b]; if (c > CAPC) c = CAPC;
  for (int t = tid; t < CAPC; t += 1024) {
    ks[t]  = (t < c) ? cscore[b * CAPC + t] : -1.f;
    kid[t] = (t < c) ? cidx[b * CAPC + t]   : 0;
  }
  __syncthreads();
  // bitonic sort, descending by score
  for (int k = 2; k <= CAPC; k <<= 1) {
    for (int j = k >> 1; j > 0; j >>= 1) {
      for (int t = tid; t < CAPC; t += 1024) {
        int p = t ^ j;
        if (p > t) {
          bool up = ((t & k) == 0);           // descending sub-sequence
          float a = ks[t], d = ks[p];
          if ((a < d) == up) {
            ks[t] = d; ks[p] = a;
            int q = kid[t]; kid[t] = kid[p]; kid[p] = q;
          }
        }
      }
      __syncthreads();
    }
  }
  float fim = (float)img[0];
  int t = tid;                                 // KPADD == blockDim
  float sc = 0.f, x1 = 0.f, y1 = 0.f, x2 = 0.f, y2 = 0.f, area = 0.f;
  int valid = 0, label = 0;
  if (t < KK && ks[t] > 0.f) {
    int idx = kid[t];
    int n = idx / CC, cl = idx % CC;
    float px = loc[2 * n], py = loc[2 * n + 1];
    const float* r = reg + ((size_t)b * NN + n) * 4;
    x1 = px - r[0]; y1 = py - r[1]; x2 = px + r[2]; y2 = py + r[3];
    x1 = fminf(fmaxf(x1, 0.f), fim); y1 = fminf(fmaxf(y1, 0.f), fim);
    x2 = fminf(fmaxf(x2, 0.f), fim); y2 = fminf(fmaxf(y2, 0.f), fim);
    float w = x2 - x1, h = y2 - y1;
    valid = (w >= MIN_SZ) && (h >= MIN_SZ);
    sc = valid ? sqrtf(ks[t]) : 0.f;
    label = cl + 1;
    area = w * h;
  }
  size_t base = (size_t)b * KPADD + t;
  boxes[base * 4 + 0] = x1; boxes[base * 4 + 1] = y1;
  boxes[base * 4 + 2] = x2; boxes[base * 4 + 3] = y2;
  ss[base] = sc; sv[base] = valid; lab[base] = label; areas[base] = area;
}

// ---------------- suppression mask: union denom via V_WMMA_F32_16X16X4_F32 -----
// D[m][n] = area_m*1 + 1*area_n  (A 16x4 = [area,1,0,0], B 4x16 = [1;area;0;0])
// Threshold test done division-free: inter*(1+T) > T*(areaSum + eps).
__global__ __launch_bounds__(256) void iou_mask_kernel(
    const float* __restrict__ boxes, const float* __restrict__ areas,
    unsigned* __restrict__ supp) {
  int b = blockIdx.x >> 6;       // 64 row-tiles per image
  int I = blockIdx.x & 63;
  int lane = threadIdx.x & 31;
  int wave = threadIdx.x >> 5;
  int n = lane & 15;
  __shared__ float rb[16 * 4];   // row boxes of tile I (shared across all J)
  __shared__ float rarea[16];
  if (threadIdx.x < 64) rb[threadIdx.x] = boxes[((size_t)b * KPADD + I * 16) * 4 + threadIdx.x];
  if (threadIdx.x < 16) rarea[threadIdx.x] = areas[b * KPADD + I * 16 + threadIdx.x];
  __syncthreads();
  v2f av;                         // A: lanes0-15 row M=lane: {K0=area, K1=1}; lanes16-31: K2/K3 = 0
  av.x = (lane < 16) ? rarea[n] : 0.f;
  av.y = (lane < 16) ? 1.f : 0.f;
  int half = (lane < 16) ? 0 : 8;
  for (int J = wave; J < 64; J += 8) {
    float areaC = areas[b * KPADD + J * 16 + n];
    v2f bv;                       // B: lanes0-15: {K0=1, K1=area_n}; lanes16-31: K2/K3 = 0
    bv.x = (lane < 16) ? 1.f : 0.f;
    bv.y = (lane < 16) ? areaC : 0.f;
    v8f cacc = {};
    v8f dsum = __builtin_amdgcn_wmma_f32_16x16x4_f32(
        false, av, false, bv, (short)0, cacc, false, false);
    int j = J * 16 + n;
    const float* bj = boxes + ((size_t)b * KPADD + j) * 4;
    float jx1 = bj[0], jy1 = bj[1], jx2 = bj[2], jy2 = bj[3];
#pragma unroll
    for (int v = 0; v < 8; ++v) {
      int m = v + half;           // C/D layout: row = v + (lane>=16 ? 8 : 0), col = lane%16
      int i = I * 16 + m;
      float w = fminf(rb[m * 4 + 2], jx2) - fmaxf(rb[m * 4 + 0], jx1);
      float h = fminf(rb[m * 4 + 3], jy2) - fmaxf(rb[m * 4 + 1], jy1);
      w = fmaxf(w, 0.f); h = fmaxf(h, 0.f);
      float inter = w * h;
      // iou > T  <=>  inter*(1+T) > T*(areaSum + eps)   (union + eps > 0 always)
      if (inter * (1.f + NMS_T) > NMS_T * (dsum[v] + 1e-9f))
        atomicOr(&supp[((size_t)b * KPADD + i) * (KPADD / 32) + (j >> 5)], 1u << (j & 31));
    }
  }
}

// ---------------- greedy NMS + top-M compaction + output -----------------------
__global__ __launch_bounds__(1024) void nms_output_kernel(
    const float* __restrict__ boxes, const float* __restrict__ ss,
    const int* __restrict__ sv, const int* __restrict__ lab,
    const unsigned* __restrict__ supp, float* __restrict__ out) {
  int b = blockIdx.x;
  int j = threadIdx.x;
  __shared__ unsigned char alive[KPADD];
  __shared__ unsigned char s_sv[KPADD];
  __shared__ float s_ss[KPADD];
  __shared__ int scan[KPADD];
  size_t base = (size_t)b * KPADD + j;
  s_ss[j] = ss[base];
  s_sv[j] = (unsigned char)sv[base];
  alive[j] = (j < KK) ? 1 : 0;
  __syncthreads();
  for (int i = 0; i < KK; ++i) {
    if (alive[i] && s_sv[i]) {
      if (j > i && j < KK) {
        unsigned w = supp[((size_t)b * KPADD + i) * (KPADD / 32) + (j >> 5)];
        if ((w >> (j & 31)) & 1u) alive[j] = 0;
      }
    }
    __syncthreads();
  }
  bool keep = (j < KK) && alive[j] && s_sv[j] && (s_ss[j] >= SCORE_T);
  scan[j] = keep ? 1 : 0;
  __syncthreads();
  for (int off = 1; off < KPADD; off <<= 1) {   // Hillis-Steele inclusive scan
    int v = scan[j];
    int add = (j >= off) ? scan[j - off] : 0;
    __syncthreads();
    scan[j] = v + add;
    __syncthreads();
  }
  int cum = scan[j];
  // output layout: boxes [B][M][4] | scores [B][M] | labels(int32) [B][M]
  float* ob  = out + (size_t)b * MM * 4;
  float* osc = out + (size_t)BB * MM * 4 + (size_t)b * MM;
  int*   olb = ((int*)out) + BB * MM * 4 + BB * MM + b * MM;
  for (int t = j; t < MM * 4; t += KPADD) ob[t] = 0.f;
  for (int t = j; t < MM; t += KPADD) { osc[t] = 0.f; olb[t] = 0; }
  __syncthreads();
  if (keep && cum <= MM) {
    int pos = cum - 1;
    const float* bb = boxes + base * 4;
    ob[pos * 4 + 0] = bb[0]; ob[pos * 4 + 1] = bb[1];
    ob[pos * 4 + 2] = bb[2]; ob[pos * 4 + 3] = bb[3];
    osc[pos] = s_ss[j];
    olb[pos] = lab[base];
  }
}

extern "C" void kernel_launch(void* const* d_in, const int* in_sizes, int n_in,
                              void* d_out, int out_size, void* d_ws, size_t ws_size,
                              hipStream_t stream) {
  const float* loc   = (const float*)d_in[0];
  const float* cls   = (const float*)d_in[1];
  const float* reg   = (const float*)d_in[2];
  const float* cness = (const float*)d_in[3];
  const int*   img   = (const int*)d_in[4];
  char* ws = (char*)d_ws;
  int*      hist   = (int*)(ws + 0);            // 16*4096*4 = 262144
  int*      cnt    = (int*)(ws + 262144);       // 64 (pad 256)
  int*      Tbin   = (int*)(ws + 262400);       // 64 (pad 256)
  float*    cscore = (float*)(ws + 262656);     // 262144
  int*      cidx   = (int*)(ws + 524800);       // 262144
  float*    boxes  = (float*)(ws + 786944);     // 16*1024*16 = 262144
  float*    ss     = (float*)(ws + 1049088);    // 65536
  int*      sv     = (int*)(ws + 1114624);      // 65536
  int*      lab    = (int*)(ws + 1180160);      // 65536
  float*    areas  = (float*)(ws + 1245696);    // 65536
  unsigned* supp   = (unsigned*)(ws + 1311232); // 16*1024*32*4 = 2097152
  float*    cnsig  = (float*)(ws + 3408384);    // 16*50000*4 = 3200000 -> total ~6.6MB
  float* out = (float*)d_out;

  init_ws<<<512, 256, 0, stream>>>(hist, cnt, supp);
  cness_kernel<<<1024, 256, 0, stream>>>(cness, cnsig);
  dim3 sg(512, BB);
  hist_kernel<<<sg, 256, 0, stream>>>(cls, cnsig, hist);
  thresh_kernel<<<BB, 32, 0, stream>>>(hist, Tbin);
  collect_kernel<<<sg, 256, 0, stream>>>(cls, cnsig, Tbin, cnt, cscore, cidx);
  sort_decode_kernel<<<BB, 1024, 0, stream>>>(cnt, cscore, cidx, loc, reg, img,
                                              boxes, ss, sv, lab, areas);
  iou_mask_kernel<<<BB * 64, 256, 0, stream>>>(boxes, areas, supp);
  nms_output_kernel<<<BB, 1024, 0, stream>>>(boxes, ss, sv, lab, supp, out);
}